// GearNet_30889404793308
// MI455X (gfx1250) — compile-verified
//
#include <hip/hip_runtime.h>
#include <hip/hip_bf16.h>

#define N_NODES  10000
#define N_EDGES  320000
#define N_REL    7
#define HID      512
#define N_GRAPHS 16
#define DIN      22
#define DH       300

typedef __attribute__((ext_vector_type(2))) float v2f;
typedef __attribute__((ext_vector_type(8))) float v8f;

// ---------------------------------------------------------------- utilities
__global__ void fill_zero(float* p, int n) {
    int i = blockIdx.x * blockDim.x + threadIdx.x;
    if (i < n) p[i] = 0.0f;
}

// ---------------------------------------------------------------- edge counts
__global__ void count_edges(const int* __restrict__ dst, const int* __restrict__ et,
                            float* __restrict__ cnt) {
    int e = blockIdx.x * blockDim.x + threadIdx.x;
    if (e < N_EDGES) atomicAdd(&cnt[et[e] * N_NODES + dst[e]], 1.0f);
}
__global__ void invert_cnt(float* c, int n) {
    int i = blockIdx.x * blockDim.x + threadIdx.x;
    if (i < n) c[i] = 1.0f / fmaxf(c[i], 1.0f);
}

// ---------------------------------------------------------------- layer 1 (Din = 22)
__global__ void scatter22(const float* __restrict__ x, const int* __restrict__ src,
                          const int* __restrict__ dst, const int* __restrict__ et,
                          float* __restrict__ aggAll /*[N][7][22]*/) {
    int e = blockIdx.x * 8 + (threadIdx.x >> 5);
    if (e >= N_EDGES) return;
    int lane = threadIdx.x & 31;
    int s = src[e], d = dst[e], r = et[e];
    if (lane < DIN) atomicAdd(&aggAll[(d * N_REL + r) * DIN + lane], x[s * DIN + lane]);
}

__global__ void layer1_root(const float* __restrict__ x, const float* __restrict__ W,
                            const float* __restrict__ b, float* __restrict__ out) {
    int idx = blockIdx.x * blockDim.x + threadIdx.x;
    if (idx >= N_NODES * HID) return;
    int v = idx / HID, c = idx % HID;
    float s = b[c];
    const float* xr = x + v * DIN;
    for (int d = 0; d < DIN; ++d) s += xr[d] * W[d * HID + c];
    out[idx] = s;
}

__global__ void layer1_rel(const float* __restrict__ aggAll, const float* __restrict__ invc,
                           const float* __restrict__ Wrel /*[7][22][512]*/,
                           float* __restrict__ out) {
    int idx = blockIdx.x * blockDim.x + threadIdx.x;
    if (idx >= N_NODES * HID) return;
    int v = idx / HID, c = idx % HID;
    float s = out[idx];
    for (int r = 0; r < N_REL; ++r) {
        float sc = invc[r * N_NODES + v];
        const float* ag = aggAll + (v * N_REL + r) * DIN;
        const float* W  = Wrel + r * DIN * HID;
        float acc = 0.0f;
        for (int d = 0; d < DIN; ++d) acc += ag[d] * W[d * HID + c];
        s += acc * sc;
    }
    out[idx] = s;
}

// ---------------------------------------------------------------- 512-wide scatter
__global__ void scatter512(const float* __restrict__ h, const int* __restrict__ src,
                           const int* __restrict__ dst, const int* __restrict__ et,
                           int rel, float* __restrict__ agg) {
    int e = blockIdx.x * 8 + (threadIdx.x >> 5);
    if (e >= N_EDGES) return;
    if (et[e] != rel) return;                 // wave-uniform
    int lane = threadIdx.x & 31;
    const float* hs = h + (size_t)src[e] * HID;
    float* ad = agg + (size_t)dst[e] * HID;
    for (int c = lane; c < HID; c += 32) atomicAdd(&ad[c], hs[c]);
}

// ---------------------------------------------------------------- FP32 WMMA GEMM
// C[M,512] (init bias | accumulate) += (rowscale * A[M,512]) @ B[512,512]
// one wave -> two adjacent 16x16 D tiles (shared A fragment -> 2 wmma per A load);
// 8 waves share two LDS-staged 512x16 B panels (64 KB).
__global__ __launch_bounds__(256) void rgcn_gemm512(
    const float* __restrict__ A, const float* __restrict__ B,
    const float* __restrict__ rowscale, const float* __restrict__ bias,
    float* __restrict__ C, int M) {
    __shared__ float Bs[2 * HID * 16];        // two 32 KB panels, row stride 16
    float* Bs0 = Bs;
    float* Bs1 = Bs + HID * 16;
    const int NB = HID / 32;                  // 16 column blocks of 32
    int nblk = blockIdx.x % NB;
    int mblk = blockIdx.x / NB;
    int tid  = threadIdx.x;
    int n0a  = nblk * 32;
    int n0b  = n0a + 16;

    // cooperative load of both B panels (float4 rows)
    for (int i = tid; i < HID * 4; i += 256) {
        int k = i >> 2, c4 = i & 3;
        const float* brow = B + (size_t)k * HID;
        ((float4*)Bs0)[i] = *(const float4*)(brow + n0a + c4 * 4);
        ((float4*)Bs1)[i] = *(const float4*)(brow + n0b + c4 * 4);
    }
    __syncthreads();

    int warp = tid >> 5, lane = tid & 31;
    int half = lane >> 4, lid = lane & 15;
    int mt = mblk * 8 + warp;
    if (mt * 16 < M) {                        // wave-uniform: EXEC all ones inside
        int m0 = mt * 16;
        v8f acc0, acc1;
        if (bias) {
            float bva = bias[n0a + lid];
            float bvb = bias[n0b + lid];
            for (int v = 0; v < 8; ++v) { acc0[v] = bva; acc1[v] = bvb; }
        } else {
            for (int v = 0; v < 8; ++v) {
                const float* crow = C + (size_t)(m0 + v + 8 * half) * HID;
                acc0[v] = crow[n0a + lid];
                acc1[v] = crow[n0b + lid];
            }
        }
        int mrow = m0 + lid;
        float scale = rowscale ? rowscale[mrow] : 1.0f;
        const float* Arow = A + (size_t)mrow * HID;
        for (int k0 = 0; k0 < HID; k0 += 4) {
            int ka = k0 + 2 * half;           // A: VGPR v -> K = v + 2*(lane>=16)
            float2 av = *(const float2*)(Arow + ka);
            v2f a; a.x = av.x * scale; a.y = av.y * scale;
            v2f b0, b1;                       // B: VGPR v -> K = v + 2*(lane>=16), N = lid
            b0.x = Bs0[ka * 16 + lid];
            b0.y = Bs0[(ka + 1) * 16 + lid];
            b1.x = Bs1[ka * 16 + lid];
            b1.y = Bs1[(ka + 1) * 16 + lid];
            acc0 = __builtin_amdgcn_wmma_f32_16x16x4_f32(
                false, a, false, b0, (short)0, acc0, false, false);
            acc1 = __builtin_amdgcn_wmma_f32_16x16x4_f32(
                false, a, false, b1, (short)0, acc1, false, false);
        }
        for (int v = 0; v < 8; ++v) {
            float* crow = C + (size_t)(m0 + v + 8 * half) * HID;
            crow[n0a + lid] = acc0[v];
            crow[n0b + lid] = acc1[v];
        }
    }
}

// ---------------------------------------------------------------- batch norm (coalesced two-phase)
__global__ __launch_bounds__(256) void bn_partial(const float* __restrict__ h, int M,
                                                  float* __restrict__ sum,
                                                  float* __restrict__ sumsq) {
    int rows_per_blk = (M + gridDim.x - 1) / gridDim.x;
    int r0 = blockIdx.x * rows_per_blk;
    int r1 = min(M, r0 + rows_per_blk);
    int c0 = threadIdx.x * 2;                 // each thread owns 2 fixed columns
    float s0 = 0.f, s1 = 0.f, q0 = 0.f, q1 = 0.f;
    for (int v = r0; v < r1; ++v) {
        float2 xv = *(const float2*)(h + (size_t)v * HID + c0);
        s0 += xv.x; q0 += xv.x * xv.x;
        s1 += xv.y; q1 += xv.y * xv.y;
    }
    atomicAdd(&sum[c0], s0);     atomicAdd(&sum[c0 + 1], s1);
    atomicAdd(&sumsq[c0], q0);   atomicAdd(&sumsq[c0 + 1], q1);
}

__global__ void bn_finalize(const float* __restrict__ sum, const float* __restrict__ sumsq,
                            int M, float* __restrict__ mu, float* __restrict__ rsig) {
    int c = blockIdx.x * blockDim.x + threadIdx.x;
    if (c >= HID) return;
    float m = sum[c] / M;
    float var = sumsq[c] / M - m * m;
    mu[c] = m;
    rsig[c] = rsqrtf(var + 1e-5f);
}

__global__ void bn_apply(float* __restrict__ h, const float* __restrict__ mu,
                         const float* __restrict__ rsig, const float* __restrict__ gamma,
                         const float* __restrict__ beta, int relu, int M) {
    int idx = blockIdx.x * blockDim.x + threadIdx.x;
    if (idx >= M * HID) return;
    int c = idx % HID;
    float y = (h[idx] - mu[c]) * rsig[c] * gamma[c] + beta[c];
    if (relu) y = fmaxf(y, 0.0f);
    h[idx] = y;
}

// ---------------------------------------------------------------- pooling + head
__global__ void pool(const float* __restrict__ h, const int* __restrict__ batch,
                     float* __restrict__ g, float* __restrict__ gcnt) {
    int v = blockIdx.x * 8 + (threadIdx.x >> 5);
    if (v >= N_NODES) return;
    int lane = threadIdx.x & 31;
    int b = batch[v];
    const float* hv = h + (size_t)v * HID;
    float* gb = g + (size_t)b * HID;
    for (int c = lane; c < HID; c += 32) atomicAdd(&gb[c], hv[c]);
    if (lane == 0) atomicAdd(&gcnt[b], 1.0f);
}

__global__ void head1(const float* __restrict__ g, const float* __restrict__ gcnt,
                      const float* __restrict__ W, const float* __restrict__ b,
                      float* __restrict__ p1) {
    int gid = blockIdx.x, c = threadIdx.x;
    if (c >= DH) return;
    float inv = 1.0f / fmaxf(gcnt[gid], 1.0f);
    float s = b[c];
    for (int k = 0; k < HID; ++k) s += g[gid * HID + k] * inv * W[k * DH + c];
    p1[gid * DH + c] = fmaxf(s, 0.0f);
}

__global__ void head2(const float* __restrict__ p1, const float* __restrict__ W,
                      const float* __restrict__ b, float* __restrict__ out) {
    int gid = blockIdx.x, c = threadIdx.x;
    if (c >= DH) return;
    float s = b[c];
    for (int k = 0; k < DH; ++k) s += p1[gid * DH + k] * W[k * DH + c];
    out[gid * DH + c] = s;
}

// ---------------------------------------------------------------- launcher
extern "C" void kernel_launch(void* const* d_in, const int* in_sizes, int n_in,
                              void* d_out, int out_size, void* d_ws, size_t ws_size,
                              hipStream_t stream) {
    const float* x      = (const float*)d_in[0];
    const int*   ei     = (const int*)d_in[1];    // [2, E]
    const int*   etype  = (const int*)d_in[2];
    const int*   batch  = (const int*)d_in[3];
    const float* W1_rel = (const float*)d_in[4];  // [7,22,512]
    const float* W1_root= (const float*)d_in[5];  // [22,512]
    const float* b1     = (const float*)d_in[6];
    const float* Wr     = (const float*)d_in[7];  // [5,7,512,512]
    const float* Wroot  = (const float*)d_in[8];  // [5,512,512]
    const float* br     = (const float*)d_in[9];  // [5,512]
    const float* gamma  = (const float*)d_in[10];
    const float* beta   = (const float*)d_in[11];
    const float* Wh1    = (const float*)d_in[12]; // [512,300]
    const float* bh1    = (const float*)d_in[13];
    const float* Wh2    = (const float*)d_in[14]; // [300,300]
    const float* bh2    = (const float*)d_in[15];
    float* out = (float*)d_out;

    const int* src = ei;
    const int* dst = ei + N_EDGES;

    // workspace carve-up
    size_t off = 0;
    auto take = [&](size_t bytes) -> float* {
        size_t o = off;
        off += (bytes + 255) & ~(size_t)255;
        return (float*)((char*)d_ws + o);
    };
    float* hA     = take((size_t)N_NODES * HID * 4);
    float* hB     = take((size_t)N_NODES * HID * 4);
    float* agg    = take((size_t)N_NODES * HID * 4);
    float* agg22  = take((size_t)N_NODES * N_REL * DIN * 4);
    float* invc   = take((size_t)N_REL * N_NODES * 4);
    float* mu     = take(HID * 4);
    float* rsig   = take(HID * 4);
    float* sumb   = take(HID * 4);
    float* sqb    = take(HID * 4);
    float* g      = take((size_t)N_GRAPHS * HID * 4);
    float* gcnt   = take(N_GRAPHS * 4);
    float* p1     = take((size_t)N_GRAPHS * DH * 4);

    const int elemNH   = N_NODES * HID;
    const int gridNH   = (elemNH + 255) / 256;
    const int gridE    = (N_EDGES + 255) / 256;
    const int gridE8   = (N_EDGES + 7) / 8;
    const int gridN8   = (N_NODES + 7) / 8;
    const int mtiles   = (N_NODES + 15) / 16;             // 625
    const int gemmGrid = ((mtiles + 7) / 8) * (HID / 32); // 79*16

    auto run_bn = [&](float* h, int relu) {
        fill_zero<<<(2 * HID + 255) / 256, 256, 0, stream>>>(sumb, 2 * HID); // sumb+sqb contiguous
        bn_partial<<<40, 256, 0, stream>>>(h, N_NODES, sumb, sqb);
        bn_finalize<<<2, 256, 0, stream>>>(sumb, sqb, N_NODES, mu, rsig);
        bn_apply<<<gridNH, 256, 0, stream>>>(h, mu, rsig, gamma, beta, relu, N_NODES);
    };

    // 1) per-relation in-degree reciprocals
    fill_zero<<<(N_REL * N_NODES + 255) / 256, 256, 0, stream>>>(invc, N_REL * N_NODES);
    count_edges<<<gridE, 256, 0, stream>>>(dst, etype, invc);
    invert_cnt<<<(N_REL * N_NODES + 255) / 256, 256, 0, stream>>>(invc, N_REL * N_NODES);

    // 2) layer 1: 22 -> 512 (scalar path, K too small for WMMA)
    fill_zero<<<(N_NODES * N_REL * DIN + 255) / 256, 256, 0, stream>>>(agg22, N_NODES * N_REL * DIN);
    scatter22<<<gridE8, 256, 0, stream>>>(x, src, dst, etype, agg22);
    layer1_root<<<gridNH, 256, 0, stream>>>(x, W1_root, b1, hA);
    layer1_rel<<<gridNH, 256, 0, stream>>>(agg22, invc, W1_rel, hA);
    run_bn(hA, 1);

    // 3) layers 2-6: 512 -> 512 via fp32 WMMA
    float* cur = hA;
    float* nxt = hB;
    for (int L = 0; L < 5; ++L) {
        rgcn_gemm512<<<gemmGrid, 256, 0, stream>>>(
            cur, Wroot + (size_t)L * HID * HID, nullptr, br + (size_t)L * HID,
            nxt, N_NODES);
        for (int r = 0; r < N_REL; ++r) {
            fill_zero<<<gridNH, 256, 0, stream>>>(agg, elemNH);
            scatter512<<<gridE8, 256, 0, stream>>>(cur, src, dst, etype, r, agg);
            rgcn_gemm512<<<gemmGrid, 256, 0, stream>>>(
                agg, Wr + ((size_t)L * N_REL + r) * HID * HID,
                invc + (size_t)r * N_NODES, nullptr, nxt, N_NODES);
        }
        run_bn(nxt, (L < 4) ? 1 : 0);
        float* t = cur; cur = nxt; nxt = t;
    }

    // 4) global mean pool + projection head
    fill_zero<<<(N_GRAPHS * HID + 255) / 256, 256, 0, stream>>>(g, N_GRAPHS * HID);
    fill_zero<<<1, 256, 0, stream>>>(gcnt, N_GRAPHS);
    pool<<<gridN8, 256, 0, stream>>>(cur, batch, g, gcnt);
    head1<<<N_GRAPHS, 320, 0, stream>>>(g, gcnt, Wh1, bh1, p1);
    head2<<<N_GRAPHS, 320, 0, stream>>>(p1, Wh2, bh2, out);
}